// Model_34746285424773
// MI455X (gfx1250) — compile-verified
//
#include <hip/hip_runtime.h>

#define N_SEG 2000
#define DDIM  240
#define LLVL  8
#define PNODE 50000

#define M_TILES   (PNODE / 16)   // 3125
#define NT_PER_WV 5              // N-tiles per wave (80 columns)
#define STRIPES   3              // 3 * 80 = 240 columns
#define STRIPE_W  (NT_PER_WV * 16)          // 80
#define LDS_STRIDE 244           // padded row stride (floats) -> conflict-free ds_load_b64

typedef __attribute__((ext_vector_type(2))) float v2f;
typedef __attribute__((ext_vector_type(8))) float v8f;

// ---------------- zero workspace ----------------
__global__ void zero_kernel(float* p, int n) {
    int i = blockIdx.x * blockDim.x + threadIdx.x;
    if (i < n) p[i] = 0.0f;
}

// ---------------- counts[seg] += end_mask over all L*P nodes ----------------
__global__ void counts_kernel(const int* __restrict__ seg_ids,
                              const int* __restrict__ end_mask,
                              float* __restrict__ counts) {
    int i = blockIdx.x * blockDim.x + threadIdx.x;
    if (i < LLVL * PNODE) {
        if (end_mask[i]) atomicAdd(&counts[seg_ids[i]], 1.0f);
    }
}

// ---------------- level 0: Z0 gather + fused masked accumulation ----------------
__global__ void level0_kernel(const float* __restrict__ feats,
                              const int* __restrict__ seg_ids0,
                              const int* __restrict__ end_mask0,
                              float* __restrict__ Z0,
                              float* __restrict__ sums) {
    int t = blockIdx.x * blockDim.x + threadIdx.x;
    if (t >= PNODE * DDIM) return;
    int i = t / DDIM;
    int j = t - i * DDIM;
    int seg = seg_ids0[i];
    float z = feats[(long)seg * DDIM + j];
    Z0[t] = z;
    if (end_mask0[i]) atomicAdd(&sums[(long)seg * DDIM + j], z);
}

// ---------------- one propagation level ----------------
// Block: 8 waves share one 80-column W-stripe staged in LDS via async-to-LDS.
// Wave: 1 M-tile x 5 N-tiles; A fragment loaded once per K-step, reused 5x.
__global__ __launch_bounds__(256)
void step_wmma_kernel(const float* __restrict__ Zprev,
                      const float* __restrict__ W,
                      const float* __restrict__ feats,
                      const float* __restrict__ attn,
                      const int*  __restrict__ seg_l,     // seg_ids[l]
                      const int*  __restrict__ seg_lm1,   // seg_ids[l-1]
                      const int*  __restrict__ par_l,     // parent_local[l]
                      const int*  __restrict__ mask_l,    // end_mask[l]
                      float* __restrict__ Zout,
                      float* __restrict__ sums) {
    __shared__ float Wlds[STRIPE_W * LDS_STRIDE];   // 80 * 244 * 4 = 78080 B

    const int stripe = blockIdx.x % STRIPES;
    const int mgroup = blockIdx.x / STRIPES;
    const int nb = stripe * STRIPE_W;

    // ---- async fill of the W stripe: B(k,j) = W[j,k]; LDS row r = column (nb+r) ----
    // 80 rows x 240 cols as float4 chunks: 80*60 = 4800 chunks
    {
        unsigned long long wbase = (unsigned long long)(W + (size_t)nb * DDIM);
        for (int chunk = threadIdx.x; chunk < STRIPE_W * (DDIM / 4); chunk += 256) {
            int r  = chunk / (DDIM / 4);
            int c  = (chunk - r * (DDIM / 4)) * 4;
            unsigned lds_addr = (unsigned)(unsigned long long)&Wlds[r * LDS_STRIDE + c];
            unsigned goff = (unsigned)((r * DDIM + c) * 4);
            asm volatile("global_load_async_to_lds_b128 %0, %1, %2"
                         :: "v"(lds_addr), "v"(goff), "s"(wbase) : "memory");
        }
        asm volatile("s_wait_asynccnt 0x0" ::: "memory");
    }
    __syncthreads();

    const int wave = threadIdx.x >> 5;
    const int lane = threadIdx.x & 31;
    const int m_tile = mgroup * 8 + wave;
    if (m_tile >= M_TILES) return;            // wave-uniform -> EXEC all-1 for WMMA

    const int m_base = m_tile * 16;
    const int half = lane >> 4;               // 0: K{0,1}, 1: K{2,3}
    const int nloc = lane & 15;

    // A: gathered Zprev row for this lane's M index
    const int parA = par_l[m_base + nloc];
    const float* aptr = Zprev + (long)parA * DDIM + 2 * half;
    // B fragments from LDS: row (t*16 + nloc), dword offset k0 + 2*half
    const float* bbase = &Wlds[nloc * LDS_STRIDE + 2 * half];

    v8f acc[NT_PER_WV];
#pragma unroll
    for (int t = 0; t < NT_PER_WV; ++t) acc[t] = (v8f){};

#pragma unroll 2
    for (int k0 = 0; k0 < DDIM; k0 += 4) {
        v2f a = *(const v2f*)(aptr + k0);
#pragma unroll
        for (int t = 0; t < NT_PER_WV; ++t) {
            v2f b = *(const v2f*)(bbase + t * (16 * LDS_STRIDE) + k0);
            acc[t] = __builtin_amdgcn_wmma_f32_16x16x4_f32(false, a, false, b,
                                                           (short)0, acc[t],
                                                           false, false);
        }
    }

    // Epilogue: D layout — VGPR r holds M=r (lanes 0-15) / M=r+8 (lanes 16-31)
    const int j0 = nb + nloc;
#pragma unroll
    for (int r = 0; r < 8; ++r) {
        int m   = m_base + half * 8 + r;
        int seg = seg_l[m];
        int ps  = seg_lm1[par_l[m]];
        float alpha = attn[(long)ps * N_SEG + seg];
        bool msk = mask_l[m] != 0;
        const float* frow = feats + (long)seg * DDIM;
        float* zrow = Zout + (long)m * DDIM;
        float* srow = sums + (long)seg * DDIM;
#pragma unroll
        for (int t = 0; t < NT_PER_WV; ++t) {
            int j = j0 + t * 16;
            float z = fmaf(alpha, acc[t][r], frow[j]);
            zrow[j] = z;
            if (msk) atomicAdd(&srow[j], z);
        }
    }
}

// ---------------- finalize: segments + z_end = sums / max(counts, 1) ----------------
__global__ void finalize_kernel(const float* __restrict__ sums,
                                const float* __restrict__ counts,
                                float* __restrict__ out) {
    int t = blockIdx.x * blockDim.x + threadIdx.x;
    if (t < N_SEG) out[t] = (float)t;  // segments = arange(n_s)
    if (t < N_SEG * DDIM) {
        int s = t / DDIM;
        out[N_SEG + t] = sums[t] / fmaxf(counts[s], 1.0f);
    }
}

extern "C" void kernel_launch(void* const* d_in, const int* in_sizes, int n_in,
                              void* d_out, int out_size, void* d_ws, size_t ws_size,
                              hipStream_t stream) {
    const float* feats = (const float*)d_in[0];  // [2000, 240]
    const float* attn  = (const float*)d_in[1];  // [2000, 2000]
    const float* W     = (const float*)d_in[2];  // [240, 240]
    const int*   segi  = (const int*)d_in[3];    // [8, 50000]
    const int*   par   = (const int*)d_in[4];    // [8, 50000]
    const int*   mask  = (const int*)d_in[5];    // [8, 50000]
    float* out = (float*)d_out;

    // workspace layout (floats): bufA | bufB | sums | counts  (~98 MB -> L2-resident)
    float* bufA   = (float*)d_ws;
    float* bufB   = bufA + (size_t)PNODE * DDIM;
    float* sums   = bufB + (size_t)PNODE * DDIM;
    float* counts = sums + (size_t)N_SEG * DDIM;

    int nz = N_SEG * DDIM + N_SEG;  // sums + counts contiguous
    zero_kernel<<<(nz + 255) / 256, 256, 0, stream>>>(sums, nz);
    counts_kernel<<<(LLVL * PNODE + 255) / 256, 256, 0, stream>>>(segi, mask, counts);
    level0_kernel<<<((size_t)PNODE * DDIM + 255) / 256, 256, 0, stream>>>(
        feats, segi, mask, bufA, sums);

    const int mgroups = (M_TILES + 7) / 8;          // 391
    const int blocks = mgroups * STRIPES;           // 1173

    float* zin = bufA;
    float* zout = bufB;
    for (int l = 1; l < LLVL; ++l) {
        step_wmma_kernel<<<blocks, 256, 0, stream>>>(
            zin, W, feats, attn,
            segi + (size_t)l * PNODE, segi + (size_t)(l - 1) * PNODE,
            par + (size_t)l * PNODE, mask + (size_t)l * PNODE,
            zout, sums);
        float* t = zin; zin = zout; zout = t;
    }

    finalize_kernel<<<(N_SEG * DDIM + 255) / 256, 256, 0, stream>>>(sums, counts, out);
}